// GaussianVariationalAnsatz_7129645711635
// MI455X (gfx1250) — compile-verified
//
#include <hip/hip_runtime.h>
#include <hip/hip_bf16.h>

// GaussianVariationalAnsatz on MI455X (gfx1250).
// cov (4096x4096 f32, 64MB) dominates: 4 layers x (row sweep + col sweep),
// ~1GB total HBM traffic => ~45us at 23.3TB/s. Staircase gates are fused
// into 16x16 group composites T_g (7 gates/group, 14-row advance) and each
// group update runs as 4 chained V_WMMA_F32_16X16X4_F32 (16x16x16 product).

#define N_MODES 2048
#define NQ      (2 * N_MODES)                 // 4096 quadratures
#define DEPTH   4
#define NSTEP   (N_MODES - 1)                 // 2047 staircase steps / layer
#define GSTEPS  7                             // gates fused per group
#define GADV    14                            // quadrature advance per group
#define NG      ((NSTEP + GSTEPS - 1) / GSTEPS)  // 293 groups
#define SQRT2F  1.41421356237309515f

typedef float v2f __attribute__((ext_vector_type(2)));
typedef float v8f __attribute__((ext_vector_type(8)));

// ---------------------------------------------------------------------------
// Phase A1: build 16x16 group composites T_g = E_{7g+6} ... E_{7g}
// (E_i = BS(theta_i)@R(phi_i) embedded at local quadrature 2*(i-7g)).
// One block per (layer, group); threads 0..15 each own one column of T.
// ---------------------------------------------------------------------------
__global__ void gva_build_T(const float* __restrict__ theta,
                            const float* __restrict__ phi,
                            float* __restrict__ Tws) {
    const int bid   = blockIdx.x;
    const int layer = bid / NG;
    const int g     = bid - layer * NG;
    const int lane  = threadIdx.x;
    if (lane >= 16) return;

    const float* th = theta + layer * NSTEP;
    const float* ph = phi   + layer * N_MODES;

    float R[16];
#pragma unroll
    for (int r = 0; r < 16; ++r) R[r] = (r == lane) ? 1.0f : 0.0f;

#pragma unroll
    for (int j = 0; j < GSTEPS; ++j) {
        const int i = g * GSTEPS + j;
        if (i < NSTEP) {
            float c, sn, ct, st;
            sincosf(ph[i], &sn, &c);
            sincosf(th[i], &st, &ct);
            // M = BS4(theta) @ R4(phi):
            //  [ ct*c  -ct*s   st   0 ]
            //  [ ct*s   ct*c   0    st]
            //  [-st*c   st*s   ct   0 ]
            //  [-st*s  -st*c   0    ct]
            const float r0 = R[2 * j + 0], r1 = R[2 * j + 1];
            const float r2 = R[2 * j + 2], r3 = R[2 * j + 3];
            R[2 * j + 0] =  ct * c * r0 - ct * sn * r1 + st * r2;
            R[2 * j + 1] =  ct * sn * r0 + ct * c * r1 + st * r3;
            R[2 * j + 2] = -st * c * r0 + st * sn * r1 + ct * r2;
            R[2 * j + 3] = -st * sn * r0 - st * c * r1 + ct * r3;
        }
    }

    float* T = Tws + (size_t)(layer * NG + g) * 256;  // row-major 16x16
#pragma unroll
    for (int r = 0; r < 16; ++r) T[r * 16 + lane] = R[r];
}

// ---------------------------------------------------------------------------
// Phase A2: squeezing scale vectors s[2i]=e^{-r_i}, s[2i+1]=e^{+r_i}.
// ---------------------------------------------------------------------------
__global__ void gva_build_s(const float* __restrict__ sq_r,
                            float* __restrict__ sws) {
    const int idx = blockIdx.x * blockDim.x + threadIdx.x;
    if (idx >= DEPTH * NQ) return;
    const int layer = idx / NQ;
    const int q     = idx - layer * NQ;
    const float r   = sq_r[layer * N_MODES + (q >> 1)];
    sws[idx] = __expf((q & 1) ? r : -r);
}

// ---------------------------------------------------------------------------
// mu evolution (independent of cov): displacement + scale + group sweep,
// all 4 layers, single block through LDS.
// ---------------------------------------------------------------------------
__global__ void gva_mu(const float* __restrict__ mu_in,
                       const float* __restrict__ alpha,
                       const float* __restrict__ sws,
                       const float* __restrict__ Tws,
                       float* __restrict__ mu_out) {
    __shared__ float m[NQ];
    const int t = threadIdx.x;
    for (int q = t; q < NQ; q += blockDim.x) m[q] = mu_in[q];
    __syncthreads();

    for (int layer = 0; layer < DEPTH; ++layer) {
        const float* s  = sws + layer * NQ;
        const float* al = alpha + layer * N_MODES;
        for (int q = t; q < NQ; q += blockDim.x) {
            float v = m[q];
            if (!(q & 1)) v += SQRT2F * al[q >> 1];   // real alpha: x only
            m[q] = v * s[q];
        }
        __syncthreads();

        const float* T = Tws + (size_t)layer * NG * 256;
        for (int g = 0; g < NG; ++g) {
            const int base = GADV * g;
            float acc = 0.0f;
            if (t < 16) {
                const float* Tg = T + g * 256 + t * 16;
#pragma unroll
                for (int k = 0; k < 16; ++k) {
                    const int q = base + k;
                    acc += Tg[k] * ((q < NQ) ? m[q] : 0.0f);
                }
            }
            __syncthreads();
            if (t < 16 && base + t < NQ) m[base + t] = acc;
            __syncthreads();
        }
    }
    for (int q = t; q < NQ; q += blockDim.x) mu_out[q] = m[q];
}

// ---------------------------------------------------------------------------
// Row sweep: cov -> P * (diag(s) applied on load) * cov.
// Each wave owns 16 columns; sweeps groups g ascending; 2 carried rows per
// group live in per-wave LDS. WMMA: D(16x16) += T[:,4k:4k+4] @ seg[4k:4k+4,:].
// In-place safe (every row read exactly once before it is written).
// ---------------------------------------------------------------------------
__global__ __launch_bounds__(256) void gva_row_sweep(const float* src,
                                                     float* dst,
                                                     const float* __restrict__ s,
                                                     const float* __restrict__ Tl) {
    __shared__ float carry[8][32];                    // [wave][col*2 + row01]
    const int lane  = threadIdx.x & 31;
    const int wslot = threadIdx.x >> 5;
    const int wv    = blockIdx.x * 8 + wslot;
    const int lo    = lane & 15;
    const int hi    = lane >> 4;
    const int col   = wv * 16 + lo;

    for (int g = 0; g < NG; ++g) {
        const int rowbase = GADV * g;
        const float* Tg = Tl + (size_t)g * 256;

        // prefetch next group's fresh rows
        if (rowbase + GADV + 2 + hi < NQ)
            __builtin_prefetch(&src[(size_t)(rowbase + GADV + 2 + hi) * NQ + col], 0, 1);

        v8f acc = {0.f, 0.f, 0.f, 0.f, 0.f, 0.f, 0.f, 0.f};
#pragma unroll
        for (int k = 0; k < 4; ++k) {
            v2f a, b;
#pragma unroll
            for (int j = 0; j < 2; ++j) {
                const int kk = 4 * k + j + 2 * hi;     // local row (K) index
                a[j] = Tg[lo * 16 + kk];               // A: T[M=lo][K=kk]
                float bv;
                if (kk < 2) {                          // carried rows 0,1 (hi==0,k==0)
                    bv = (g == 0) ? src[(size_t)kk * NQ + col] * s[kk]
                                  : carry[wslot][lo * 2 + kk];
                } else {
                    const int row = rowbase + kk;
                    bv = (row < NQ) ? src[(size_t)row * NQ + col] * s[row] : 0.0f;
                }
                b[j] = bv;
            }
            acc = __builtin_amdgcn_wmma_f32_16x16x4_f32(
                false, a, false, b, (short)0, acc, false, false);
        }

        // carry rows 14,15 (D rows 14,15 = acc[6],acc[7] on lanes 16..31)
        if (hi) {
            carry[wslot][lo * 2 + 0] = acc[6];
            carry[wslot][lo * 2 + 1] = acc[7];
        }
        // store finalized rows 0..13
#pragma unroll
        for (int v = 0; v < 8; ++v) {
            const int M = v + 8 * hi;
            const int row = rowbase + M;
            if (M < GADV && row < NQ) dst[(size_t)row * NQ + col] = acc[v];
        }
        __syncthreads();
    }
}

// ---------------------------------------------------------------------------
// Column sweep: X -> (X * diag(s) applied on load) * P^T, in place.
// Each wave owns 16 rows; 2 carried columns per group in per-wave LDS.
// WMMA: D(16x16) += seg[:,4k:4k+4] @ (T^T)[4k:4k+4,:]  (B = T transposed).
// ---------------------------------------------------------------------------
__global__ __launch_bounds__(256) void gva_col_sweep(float* X,
                                                     const float* __restrict__ s,
                                                     const float* __restrict__ Tl) {
    __shared__ float carry[8][32];                    // [wave][row*2 + col01]
    const int lane  = threadIdx.x & 31;
    const int wslot = threadIdx.x >> 5;
    const int wv    = blockIdx.x * 8 + wslot;
    const int lo    = lane & 15;
    const int hi    = lane >> 4;
    const int rowq  = wv * 16 + lo;

    for (int g = 0; g < NG; ++g) {
        const int colbase = GADV * g;
        const float* Tg = Tl + (size_t)g * 256;

        if (colbase + GADV + 2 < NQ)
            __builtin_prefetch(&X[(size_t)rowq * NQ + colbase + GADV + 2], 0, 1);

        v8f acc = {0.f, 0.f, 0.f, 0.f, 0.f, 0.f, 0.f, 0.f};
#pragma unroll
        for (int k = 0; k < 4; ++k) {
            v2f a, b;
#pragma unroll
            for (int j = 0; j < 2; ++j) {
                const int kk = 4 * k + j + 2 * hi;     // local column (K) index
                float av;
                if (kk < 2) {                          // carried cols 0,1
                    av = (g == 0) ? X[(size_t)rowq * NQ + kk] * s[kk]
                                  : carry[wslot][lo * 2 + kk];
                } else {
                    const int cl = colbase + kk;
                    av = (cl < NQ) ? X[(size_t)rowq * NQ + cl] * s[cl] : 0.0f;
                }
                a[j] = av;                             // A: X[M=lo][K=kk]
                b[j] = Tg[lo * 16 + kk];               // B: T^T[K][N=lo] = T[lo][kk]
            }
            acc = __builtin_amdgcn_wmma_f32_16x16x4_f32(
                false, a, false, b, (short)0, acc, false, false);
        }

        // carry columns 14,15 (D cols N=14,15 live on lanes 14,15,30,31)
        if (lo >= GADV) {
#pragma unroll
            for (int v = 0; v < 8; ++v)
                carry[wslot][(v + 8 * hi) * 2 + (lo - GADV)] = acc[v];
        }
        // store finalized columns 0..13
        if (lo < GADV) {
            const int cl = colbase + lo;
            if (cl < NQ) {
#pragma unroll
                for (int v = 0; v < 8; ++v)
                    X[(size_t)(wv * 16 + v + 8 * hi) * NQ + cl] = acc[v];
            }
        }
        __syncthreads();
    }
}

// ---------------------------------------------------------------------------
extern "C" void kernel_launch(void* const* d_in, const int* in_sizes, int n_in,
                              void* d_out, int out_size, void* d_ws, size_t ws_size,
                              hipStream_t stream) {
    const float* mu_in  = (const float*)d_in[0];  // (2n,)
    const float* cov_in = (const float*)d_in[1];  // (2n,2n)
    const float* alpha  = (const float*)d_in[2];  // (d,n)
    const float* sq_r   = (const float*)d_in[3];  // (d,n)
    const float* theta  = (const float*)d_in[4];  // (d,n-1)
    const float* phi    = (const float*)d_in[5];  // (d,n)

    float* out_mu  = (float*)d_out;               // [0, NQ)
    float* out_cov = out_mu + NQ;                 // [NQ, NQ + NQ*NQ)

    float* Tws = (float*)d_ws;                    // DEPTH*NG*256 floats (~1.2MB)
    float* sws = Tws + (size_t)DEPTH * NG * 256;  // DEPTH*NQ floats

    // Phase A: parameter-only precompute (composites + scales)
    gva_build_T<<<DEPTH * NG, 32, 0, stream>>>(theta, phi, Tws);
    gva_build_s<<<(DEPTH * NQ + 255) / 256, 256, 0, stream>>>(sq_r, sws);

    // mu path (independent of cov)
    gva_mu<<<1, 256, 0, stream>>>(mu_in, alpha, sws, Tws, out_mu);

    // cov path: per layer, fused-scale row sweep then column sweep
    for (int layer = 0; layer < DEPTH; ++layer) {
        const float* sl = sws + (size_t)layer * NQ;
        const float* Tl = Tws + (size_t)layer * NG * 256;
        const float* src = (layer == 0) ? cov_in : out_cov;
        gva_row_sweep<<<NQ / (16 * 8), 256, 0, stream>>>(src, out_cov, sl, Tl);
        gva_col_sweep<<<NQ / (16 * 8), 256, 0, stream>>>(out_cov, sl, Tl);
    }
}